// GeneGraphProjection_28114855920011
// MI455X (gfx1250) — compile-verified
//
#include <hip/hip_runtime.h>

// ---------------------------------------------------------------------------
// 2-layer GCN (PyG GCNConv semantics) for MI455X / gfx1250.
//   layer: h = x @ W ; out[dst] += dinv[src]*ew*dinv[dst] * h[src] ; + self
//          loop (norm = dinv^2) ; + bias ; relu between layers (fused).
// GEMMs use V_WMMA_F32_16X16X4_F32 (full f32 precision, matches reference).
// Scatter uses hardware global_atomic_add_f32 (L2-resident).
// ---------------------------------------------------------------------------

#define N_NODES 50000
#define N_EDGES 1000000
#define DIM 256

#define MT 5                       // 16-row M tiles per block
#define ROWS_PER_BLOCK (16 * MT)   // 80 rows ; 50000 / 80 = 625 exactly
#define GEMM_BLOCKS (N_NODES / ROWS_PER_BLOCK)

typedef __attribute__((ext_vector_type(2))) float v2f;
typedef __attribute__((ext_vector_type(8))) float v8f;

// ---------------- degree / normalization ----------------

__global__ void deg_init_kernel(float* __restrict__ deg) {
    int i = blockIdx.x * blockDim.x + threadIdx.x;
    if (i < N_NODES) deg[i] = 1.0f;               // self-loop weight = 1
}

__global__ void deg_edges_kernel(const int* __restrict__ dst,
                                 const float* __restrict__ ew,
                                 float* __restrict__ deg) {
    int e = blockIdx.x * blockDim.x + threadIdx.x;
    if (e < N_EDGES) unsafeAtomicAdd(&deg[dst[e]], ew[e]);
}

__global__ void dinv_kernel(const float* __restrict__ deg,
                            float* __restrict__ dinv) {
    int i = blockIdx.x * blockDim.x + threadIdx.x;
    if (i < N_NODES) {
        float d = deg[i];
        dinv[i] = (d > 0.0f) ? rsqrtf(d) : 0.0f;
    }
}

// ---------------- dense GEMM: Y = (relu?)(X) @ W  via f32 WMMA ----------------
// wave32: one wave owns a 16x16 output tile; 16 waves/block cover N=256;
// each wave accumulates MT M-tiles so a B fragment is reused MT times.
//
// V_WMMA_F32_16X16X4_F32 operand layouts (ISA 7.12.2, 32-bit):
//   A (16x4): lane L(0-15) holds row M=L, {v0,v1}={K0,K1}; lane L+16: {K2,K3}
//   B (4x16): lane L(0-15) holds col N=L, {v0,v1}={K0,K1}; lane L+16: {K2,K3}
//   C/D:      vgpr r: lanes 0-15 -> (M=r, N=lane); lanes 16-31 -> (M=r+8)

template <bool RELU_IN>
__global__ __launch_bounds__(512) void gcn_gemm_kernel(
    const float* __restrict__ X, const float* __restrict__ W,
    float* __restrict__ Y) {
    const int lane = threadIdx.x & 31;
    const int warp = threadIdx.x >> 5;     // 0..15 -> N tile
    const int n0   = warp * 16;
    const int m0   = blockIdx.x * ROWS_PER_BLOCK;
    const int lmod = lane & 15;
    const int lhi  = lane >> 4;            // selects K pair {0,1} vs {2,3}

    v8f acc[MT];
#pragma unroll
    for (int t = 0; t < MT; ++t) acc[t] = (v8f){0, 0, 0, 0, 0, 0, 0, 0};

    for (int k0 = 0; k0 < DIM; k0 += 4) {
        const int ka = k0 + 2 * lhi;
        v2f b;
        b.x = W[(size_t)ka * DIM + n0 + lmod];
        b.y = W[(size_t)(ka + 1) * DIM + n0 + lmod];
#pragma unroll
        for (int t = 0; t < MT; ++t) {
            const float* xr = X + (size_t)(m0 + t * 16 + lmod) * DIM + ka;
            v2f a;
            a.x = xr[0];
            a.y = xr[1];
            if (RELU_IN) {
                a.x = fmaxf(a.x, 0.0f);
                a.y = fmaxf(a.y, 0.0f);
            }
            acc[t] = __builtin_amdgcn_wmma_f32_16x16x4_f32(
                false, a, false, b, (short)0, acc[t], false, false);
        }
    }

    const int ncol = n0 + lmod;
    const int mb   = 8 * lhi;
#pragma unroll
    for (int t = 0; t < MT; ++t)
#pragma unroll
        for (int r = 0; r < 8; ++r)
            Y[(size_t)(m0 + t * 16 + mb + r) * DIM + ncol] = acc[t][r];
}

// ---------------- aggregation ----------------
// init: out[n][c] = bias[c] + dinv[n]^2 * h[n][c]   (self-loop + bias)
__global__ void agg_init_kernel(const float* __restrict__ h,
                                const float* __restrict__ dinv,
                                const float* __restrict__ bias,
                                float* __restrict__ out) {
    int row = blockIdx.x;
    int c   = threadIdx.x;
    float s = dinv[row];
    size_t idx = (size_t)row * DIM + c;
    out[idx] = bias[c] + s * s * h[idx];
}

// edges: out[dst][c] += dinv[src]*ew*dinv[dst] * h[src][c]
// one edge per 256-thread block: uniform edge scalars, coalesced row atomics
__global__ void agg_edges_kernel(const int* __restrict__ src,
                                 const int* __restrict__ dst,
                                 const float* __restrict__ ew,
                                 const float* __restrict__ dinv,
                                 const float* __restrict__ h,
                                 float* __restrict__ out) {
    int e = blockIdx.x;
    int s = src[e];
    int d = dst[e];
    float norm = dinv[s] * ew[e] * dinv[d];
    int c = threadIdx.x;
    float v = norm * h[(size_t)s * DIM + c];
    unsafeAtomicAdd(&out[(size_t)d * DIM + c], v);
}

// ---------------- launch ----------------

extern "C" void kernel_launch(void* const* d_in, const int* in_sizes, int n_in,
                              void* d_out, int out_size, void* d_ws,
                              size_t ws_size, hipStream_t stream) {
    (void)in_sizes; (void)n_in; (void)out_size; (void)ws_size;

    const float* x  = (const float*)d_in[0];
    const int*   ei = (const int*)d_in[1];     // (2, E) flat: src then dst
    const float* ew = (const float*)d_in[2];
    const float* W1 = (const float*)d_in[3];
    const float* b1 = (const float*)d_in[4];
    const float* W2 = (const float*)d_in[5];
    const float* b2 = (const float*)d_in[6];

    const int* src = ei;
    const int* dst = ei + N_EDGES;

    float* deg  = (float*)d_ws;                       // 50000
    float* dinv = deg + N_NODES;                      // 50000
    float* hA   = dinv + N_NODES;                     // 50000*256
    float* hB   = hA + (size_t)N_NODES * DIM;         // 50000*256
    float* out  = (float*)d_out;

    const int T = 256;

    // shared symmetric normalization for both layers
    deg_init_kernel<<<(N_NODES + T - 1) / T, T, 0, stream>>>(deg);
    deg_edges_kernel<<<(N_EDGES + T - 1) / T, T, 0, stream>>>(dst, ew, deg);
    dinv_kernel<<<(N_NODES + T - 1) / T, T, 0, stream>>>(deg, dinv);

    // layer 1: hA = x @ W1 ; hB = scatter(hA) + b1   (relu deferred)
    gcn_gemm_kernel<false><<<GEMM_BLOCKS, 512, 0, stream>>>(x, W1, hA);
    agg_init_kernel<<<N_NODES, T, 0, stream>>>(hA, dinv, b1, hB);
    agg_edges_kernel<<<N_EDGES, T, 0, stream>>>(src, dst, ew, dinv, hA, hB);

    // layer 2: hA = relu(hB) @ W2 ; out = scatter(hA) + b2
    gcn_gemm_kernel<true><<<GEMM_BLOCKS, 512, 0, stream>>>(hB, W2, hA);
    agg_init_kernel<<<N_NODES, T, 0, stream>>>(hA, dinv, b2, out);
    agg_edges_kernel<<<N_EDGES, T, 0, stream>>>(src, dst, ew, dinv, hA, out);
}